// RadarGATAggregator_21002390077952
// MI455X (gfx1250) — compile-verified
//
#include <hip/hip_runtime.h>
#include <math.h>

// Problem constants (fixed by the reference)
#define Bn   4
#define Nn   10000
#define En   160000
#define EFn  170000          // E + N (self loops appended)
#define INn  64
#define HCn  256             // H * OUT
#define SLOPE 0.2f

typedef __attribute__((ext_vector_type(2))) float v2f;
typedef __attribute__((ext_vector_type(8))) float v8f;

// ---------------- init kernels ----------------
__global__ void k_init_node(float* deg, float* sumattr, float* maxbuf, float* denom) {
    int i = blockIdx.x * blockDim.x + threadIdx.x;
    if (i < Nn) { deg[i] = 0.f; sumattr[i] = 0.f; }
    if (i < Bn * Nn * 2) { maxbuf[i] = -3.402823466e38f; denom[i] = 0.f; }
}

__global__ void k_init_out(float* __restrict__ out, const float* __restrict__ bias) {
    size_t i = (size_t)blockIdx.x * blockDim.x + threadIdx.x;
    if (i < (size_t)Bn * Nn * HCn) out[i] = bias[i & 255u];
}

// ---------------- self-loop attr (mean of incoming edge_attr) ----------------
__global__ void k_edge_accum(const int* __restrict__ dst, const float* __restrict__ edge_attr,
                             float* deg, float* sumattr) {
    int e = blockIdx.x * blockDim.x + threadIdx.x;
    if (e < En) {
        int d = dst[e];
        atomicAdd(&deg[d], 1.0f);
        atomicAdd(&sumattr[d], edge_attr[e]);
    }
}

__global__ void k_loop_attr(const float* deg, const float* sumattr, float* loop_attr) {
    int n = blockIdx.x * blockDim.x + threadIdx.x;
    if (n < Nn) loop_attr[n] = sumattr[n] / fmaxf(deg[n], 1.0f);
}

// ---------------- WMMA f32 GEMM: Y[M,256] = X[M,64] @ W[64,256] + bias ----------------
// One wave per 16x16 output tile; V_WMMA_F32_16X16X4_F32 along K=64 (16 steps).
// A (16x4 f32): lanes 0-15 hold M rows, VGPR0/1 = K{0,1} (lane<16) or K{2,3} (lane>=16).
// C/D (16x16 f32): 8 VGPRs, lane<16 -> M=v, lane>=16 -> M=v+8; N = lane&15.
__global__ void k_gemm(const float* __restrict__ X, const float* __restrict__ W,
                       const float* __restrict__ bias, float* __restrict__ Y) {
    int mt = blockIdx.x;            // M/16 tiles
    int nt = blockIdx.y;            // 256/16 = 16 tiles
    int lane  = threadIdx.x & 31;
    int mrow  = lane & 15;
    int khalf = lane >> 4;          // 0 or 1

    const float* Xr = X + (size_t)(mt * 16 + mrow) * INn;
    v8f acc = {};
#pragma unroll
    for (int k0 = 0; k0 < INn; k0 += 4) {
        v2f a, b;
        a.x = Xr[k0 + 2 * khalf + 0];
        a.y = Xr[k0 + 2 * khalf + 1];
        b.x = W[(size_t)(k0 + 2 * khalf + 0) * HCn + nt * 16 + mrow];
        b.y = W[(size_t)(k0 + 2 * khalf + 1) * HCn + nt * 16 + mrow];
        acc = __builtin_amdgcn_wmma_f32_16x16x4_f32(
            /*neg_a=*/false, a, /*neg_b=*/false, b,
            /*c_mod=*/(short)0, acc, /*reuse_a=*/false, /*reuse_b=*/false);
    }
    int ncol = nt * 16 + mrow;
    float bb = bias[ncol];
#pragma unroll
    for (int v = 0; v < 8; ++v) {
        int m = mt * 16 + khalf * 8 + v;
        Y[(size_t)m * HCn + ncol] = acc[v] + bb;
    }
}

// ---------------- pass 1: per-(b,edge) logits + segment max ----------------
// One wave per (b, e). Lane L covers channels [8L, 8L+8); lanes 0-15 = head 0,
// lanes 16-31 = head 1 (flat att/We are indexed directly by channel).
__global__ void k_logits(const float* __restrict__ xl, const float* __restrict__ xr,
                         const int* __restrict__ srcA, const int* __restrict__ dstA,
                         const float* __restrict__ edge_attr, const float* __restrict__ loop_attr,
                         const float* __restrict__ We, const float* __restrict__ att,
                         float* __restrict__ eL, float* __restrict__ maxbuf) {
    int wave = (blockIdx.x * blockDim.x + threadIdx.x) >> 5;
    int lane = threadIdx.x & 31;
    if (wave >= Bn * EFn) return;
    int b = wave / EFn;
    int e = wave - b * EFn;
    int s, d; float attr;
    if (e < En) { s = srcA[e]; d = dstA[e]; attr = edge_attr[e]; }
    else        { s = e - En;  d = s;       attr = loop_attr[s]; }

    const float* xls = xl + ((size_t)b * Nn + s) * HCn;
    const float* xrd = xr + ((size_t)b * Nn + d) * HCn;
    int ch0 = lane * 8;
    float p = 0.f;
#pragma unroll
    for (int j = 0; j < 8; ++j) {
        int ch = ch0 + j;
        float v = xls[ch] + xrd[ch] + attr * We[ch];
        v = v > 0.f ? v : SLOPE * v;
        p += att[ch] * v;
    }
    // reduce within each 16-lane half (one head per half)
#pragma unroll
    for (int off = 1; off < 16; off <<= 1) p += __shfl_xor(p, off, 32);
    if ((lane & 15) == 0) {
        int h = lane >> 4;
        eL[(size_t)wave * 2 + h] = p;
        atomicMax(&maxbuf[((size_t)b * Nn + d) * 2 + h], p);
    }
}

// ---------------- pass 2: exp + segment sum ----------------
__global__ void k_expsum(const int* __restrict__ dstA, const float* __restrict__ maxbuf,
                         float* __restrict__ eL, float* __restrict__ denom) {
    int i = blockIdx.x * blockDim.x + threadIdx.x;       // over B*EF*H
    if (i >= Bn * EFn * 2) return;
    int h  = i & 1;
    int be = i >> 1;
    int b  = be / EFn;
    int e  = be - b * EFn;
    int d  = (e < En) ? dstA[e] : (e - En);
    float m = maxbuf[((size_t)b * Nn + d) * 2 + h];
    float v = expf(eL[i] - m);
    eL[i] = v;
    atomicAdd(&denom[((size_t)b * Nn + d) * 2 + h], v);
}

// ---------------- pass 3: alpha * xl[src] scatter-add into out ----------------
__global__ void k_aggregate(const float* __restrict__ xl,
                            const int* __restrict__ srcA, const int* __restrict__ dstA,
                            const float* __restrict__ eL, const float* __restrict__ denom,
                            float* __restrict__ out) {
    int wave = (blockIdx.x * blockDim.x + threadIdx.x) >> 5;
    int lane = threadIdx.x & 31;
    if (wave >= Bn * EFn) return;
    int b = wave / EFn;
    int e = wave - b * EFn;
    int s, d;
    if (e < En) { s = srcA[e]; d = dstA[e]; }
    else        { s = e - En;  d = s;       }

    float a0 = eL[(size_t)wave * 2 + 0] / denom[((size_t)b * Nn + d) * 2 + 0];
    float a1 = eL[(size_t)wave * 2 + 1] / denom[((size_t)b * Nn + d) * 2 + 1];
    const float* xls = xl + ((size_t)b * Nn + s) * HCn;
    float* outd      = out + ((size_t)b * Nn + d) * HCn;
    int ch0 = lane * 8;
    float alpha = (ch0 < 128) ? a0 : a1;   // each lane's 8 channels are within one head
#pragma unroll
    for (int j = 0; j < 8; ++j) {
        int ch = ch0 + j;
        atomicAdd(&outd[ch], alpha * xls[ch]);
    }
}

extern "C" void kernel_launch(void* const* d_in, const int* in_sizes, int n_in,
                              void* d_out, int out_size, void* d_ws, size_t ws_size,
                              hipStream_t stream) {
    const float* node_feats = (const float*)d_in[0];   // [B,N,64]
    const int*   edge_index = (const int*)  d_in[1];   // [2,E]
    const float* edge_attr  = (const float*)d_in[2];   // [E,1]
    const float* Wl   = (const float*)d_in[3];
    const float* bl   = (const float*)d_in[4];
    const float* Wr   = (const float*)d_in[5];
    const float* br   = (const float*)d_in[6];
    const float* We   = (const float*)d_in[7];         // [1,256] -> flat [256]
    const float* att  = (const float*)d_in[8];         // [2,128] -> flat [256]
    const float* bias = (const float*)d_in[9];
    float* out = (float*)d_out;

    const int* srcA = edge_index;
    const int* dstA = edge_index + En;

    // workspace carve-out (floats)
    float* ws       = (float*)d_ws;
    float* xl       = ws;                                   // B*N*256 = 10.24M
    float* xr       = xl  + (size_t)Bn * Nn * HCn;          // 10.24M
    float* eL       = xr  + (size_t)Bn * Nn * HCn;          // B*EF*2 = 1.36M
    float* maxbuf   = eL  + (size_t)Bn * EFn * 2;           // B*N*2 = 80k
    float* denom    = maxbuf + (size_t)Bn * Nn * 2;         // 80k
    float* deg      = denom  + (size_t)Bn * Nn * 2;         // N
    float* sumattr  = deg + Nn;                             // N
    float* loop_attr= sumattr + Nn;                         // N

    k_init_node<<<(Bn * Nn * 2 + 255) / 256, 256, 0, stream>>>(deg, sumattr, maxbuf, denom);
    k_init_out<<<(int)(((size_t)Bn * Nn * HCn + 255) / 256), 256, 0, stream>>>(out, bias);
    k_edge_accum<<<(En + 255) / 256, 256, 0, stream>>>(dstA, edge_attr, deg, sumattr);
    k_loop_attr<<<(Nn + 255) / 256, 256, 0, stream>>>(deg, sumattr, loop_attr);

    dim3 gg(Bn * Nn / 16, HCn / 16);   // 2500 x 16 tiles, 1 wave each
    k_gemm<<<gg, 32, 0, stream>>>(node_feats, Wl, bl, xl);
    k_gemm<<<gg, 32, 0, stream>>>(node_feats, Wr, br, xr);

    int waves = Bn * EFn;              // 680000 waves, 8 per block
    k_logits<<<(waves * 32 + 255) / 256, 256, 0, stream>>>(
        xl, xr, srcA, dstA, edge_attr, loop_attr, We, att, eL, maxbuf);
    k_expsum<<<(Bn * EFn * 2 + 255) / 256, 256, 0, stream>>>(dstA, maxbuf, eL, denom);
    k_aggregate<<<(waves * 32 + 255) / 256, 256, 0, stream>>>(xl, srcA, dstA, eL, denom, out);
}